// ELLA_8057358648284
// MI455X (gfx1250) — compile-verified
//
#include <hip/hip_runtime.h>

// ELLA kernel for MI455X (gfx1250): fp32 WMMA GEMMs fed by async global->LDS copies.
//
//   phi[d,m,k] = sum_s Jz[m,d,s] * v[s,k]            (5120x16384 @ 16384x1024)
//   tmp[d,m,g] = sum_k phi[d,m,k] * G_inv[k,g]       (5120x1024  @ 1024x1024)
//   var[n,a,b] = sum_g tmp[a,n,g] * phi[b,n,g]       (diagonal of K_test only)
//   mean       = X_test @ W_net
// out = [mean (512*10) | var (512*10*10)] fp32.

typedef __attribute__((ext_vector_type(2))) float v2f;
typedef __attribute__((ext_vector_type(8))) float v8f;

#define M_PTS 512
#define D_CLS 10
#define S_DIM 16384
#define K_DIM 1024
#define DIN_  784
#define MT    (M_PTS * D_CLS)   // 5120 GEMM rows

constexpr int BM = 128, BN = 128, KT = 16;
constexpr int ASTR  = 20;            // LDS floats per A row   [BM][ASTR]  (frag-conflict-free)
constexpr int BSTRT = 17;            // LDS floats per B *col* [BN][BSTRT] (17 coprime 64 banks)
constexpr int SA    = BM * ASTR;     // 2560 floats
constexpr int SB    = BN * BSTRT;    // 2176 floats (B stored transposed: [n][k])
constexpr int BUF   = SA + SB;       // floats per double-buffer slot (18944 B)

// ---- CDNA5 async global->LDS copies (ASYNCcnt-tracked, no VGPR staging) ----
__device__ __forceinline__ void async_b128(unsigned lds_byte, const float* g) {
  asm volatile("global_load_async_to_lds_b128 %0, %1, off"
               :: "v"(lds_byte), "v"(g) : "memory");
}
__device__ __forceinline__ void async_b32(unsigned lds_byte, const float* g) {
  asm volatile("global_load_async_to_lds_b32 %0, %1, off"
               :: "v"(lds_byte), "v"(g) : "memory");
}
__device__ __forceinline__ void wait_async10() {   // previous 10-op batch retired
  asm volatile("s_wait_asynccnt 0xa" ::: "memory");
}
__device__ __forceinline__ void wait_async0() {
  asm volatile("s_wait_asynccnt 0x0" ::: "memory");
}

// C[r,c] = sum_k A[rowmap(r),k] * B[k,c]; A row stride == Kd, B row stride == 1024.
// permRows: rowmap(r) = (r%512)*10 + r/512   (Jz 'mds' -> phi 'dm' row permutation)
__global__ __launch_bounds__(256) void gemm_wmma_f32(
    const float* __restrict__ A, const float* __restrict__ B,
    float* __restrict__ C, int Kd, int permRows)
{
  __shared__ __align__(16) float smem[2 * BUF];   // ~37.9 KB, double-buffered

  const int tid  = threadIdx.x;
  const int lane = tid & 31;
  const int wid  = tid >> 5;
  const int l16  = lane & 15;
  const int half = lane >> 4;      // 0: lanes 0-15, 1: lanes 16-31
  const int wm   = wid & 1;        // 2 wave-rows * 64
  const int wn   = wid >> 1;       // 4 wave-cols * 32

  const int rowBlk = blockIdx.x * BM;
  const int colBlk = blockIdx.y * BN;

  // generic->LDS offset: low 32 bits of a __shared__ generic pointer are the
  // wave-relative LDS byte address (flat-LDS aperture rule).
  const unsigned ldsBase = (unsigned)(unsigned long long)(&smem[0]);
  const unsigned base0 = ldsBase;
  const unsigned base1 = ldsBase + BUF * 4;

  // ---- per-thread async-copy descriptors ----
  // A: 2x b128. A tile BM x KT row-major: float4 id f -> row = f>>2, kq = f&3
  const float* aCur[2];
  unsigned aOff[2];
#pragma unroll
  for (int u = 0; u < 2; ++u) {
    int f   = tid + u * 256;
    int row = f >> 2, kq = f & 3;
    int r   = rowBlk + row;
    long long rb = permRows ? (long long)((r & (M_PTS - 1)) * D_CLS + (r >> 9))
                            : (long long)r;
    aCur[u] = A + rb * (long long)Kd + kq * 4;
    aOff[u] = (unsigned)(row * ASTR + kq * 4) * 4u;
  }
  // B: 8x b32 with per-lane transpose. element e = tid + j*256:
  //   n = tid & 127 (const), k = (tid>>7) + 2*j  -> global B[k0+k][colBlk+n]
  //   LDS dest: sBt[n*BSTRT + k]  (transposed, frag {B[k][n],B[k+1][n]} contiguous)
  const int bN  = tid & 127;
  const int bKb = tid >> 7;                 // 0 or 1
  const float* bCur = B + (long long)bKb * 1024 + colBlk + bN;
  const unsigned bOff = (unsigned)(SA + bN * BSTRT + bKb) * 4u;

  auto issue = [&](unsigned bufBase) {
#pragma unroll
    for (int u = 0; u < 2; ++u) { async_b128(bufBase + aOff[u], aCur[u]); aCur[u] += KT; }
#pragma unroll
    for (int j = 0; j < 8; ++j)             // k advances by 2 per rep
      async_b32(bufBase + bOff + (unsigned)j * 8u, bCur + (long long)j * 2048);
    bCur += (long long)KT * 1024;
  };

  v8f acc[4][2];
#pragma unroll
  for (int tm = 0; tm < 4; ++tm)
#pragma unroll
    for (int tn = 0; tn < 2; ++tn) {
      v8f z = {0.f, 0.f, 0.f, 0.f, 0.f, 0.f, 0.f, 0.f};
      acc[tm][tn] = z;
    }

  auto compute = [&](const float* sbase) {
    const float* sAb = sbase;
    const float* sBt = sbase + SA;
#pragma unroll
    for (int k4 = 0; k4 < KT; k4 += 4) {
      const int kk = k4 + 2 * half;      // per-lane K of the 16x4 f32 fragment
      v2f af[4], bf[2];
#pragma unroll
      for (int tm = 0; tm < 4; ++tm) {
        const float* p = sAb + (wm * 64 + tm * 16 + l16) * ASTR + kk;
        af[tm].x = p[0];
        af[tm].y = p[1];
      }
#pragma unroll
      for (int tn = 0; tn < 2; ++tn) {
        const float* p = sBt + (wn * 32 + tn * 16 + l16) * BSTRT + kk;
        bf[tn].x = p[0];
        bf[tn].y = p[1];
      }
#pragma unroll
      for (int tm = 0; tm < 4; ++tm)
#pragma unroll
        for (int tn = 0; tn < 2; ++tn)
          acc[tm][tn] = __builtin_amdgcn_wmma_f32_16x16x4_f32(
              false, af[tm], false, bf[tn], (short)0, acc[tm][tn],
              false, false);
    }
  };

  // ---- double-buffered main loop (Kd % 32 == 0 for both GEMMs) ----
  issue(base0);                              // prefetch k-step 0 into buf0
  const int kIters = Kd / KT;                // even
  for (int it = 0; it < kIters; it += 2) {
    __syncthreads();                         // all waves done reading buf1
    if (it + 1 < kIters) { issue(base1); wait_async10(); } else { wait_async0(); }
    __syncthreads();                         // buf0 complete in every wave
    compute(smem);

    __syncthreads();                         // all waves done reading buf0
    if (it + 2 < kIters) { issue(base0); wait_async10(); } else { wait_async0(); }
    __syncthreads();                         // buf1 complete in every wave
    compute(smem + BUF);
  }

  // epilogue: C/D layout -> VGPR i holds M = i + 8*half, N = l16
#pragma unroll
  for (int tm = 0; tm < 4; ++tm)
#pragma unroll
    for (int tn = 0; tn < 2; ++tn) {
      int col = colBlk + wn * 32 + tn * 16 + l16;
#pragma unroll
      for (int i = 0; i < 8; ++i) {
        int row = rowBlk + wm * 64 + tm * 16 + i + 8 * half;
        C[(long long)row * 1024 + col] = acc[tm][tn][i];
      }
    }
}

// var[n,a,b] = sum_g tmp[a*512+n, g] * phi[b*512+n, g]
__global__ __launch_bounds__(256) void var_diag_kernel(
    const float* __restrict__ phi, const float* __restrict__ tmp,
    float* __restrict__ varOut)
{
  __shared__ float sP[D_CLS * K_DIM];        // 40 KB: phi rows of the 10 classes
  const int n   = blockIdx.x;
  const int tid = threadIdx.x;

  for (int idx = tid; idx < D_CLS * K_DIM; idx += 256) {
    int a = idx >> 10, g = idx & (K_DIM - 1);
    sP[idx] = phi[((long long)a * M_PTS + n) * K_DIM + g];
  }
  __syncthreads();

  const int wid = tid >> 5, lane = tid & 31;
  for (int p = wid; p < D_CLS * D_CLS; p += 8) {
    int a = p / D_CLS, b = p - a * D_CLS;
    const float* trow = tmp + ((long long)a * M_PTS + n) * K_DIM;
    float s = 0.f;
    for (int g = lane; g < K_DIM; g += 32)
      s += trow[g] * sP[b * K_DIM + g];
#pragma unroll
    for (int off = 16; off > 0; off >>= 1)
      s += __shfl_down(s, off, 32);
    if (lane == 0)
      varOut[(long long)n * (D_CLS * D_CLS) + p] = s;   // out[n,a,b]
  }
}

__global__ __launch_bounds__(256) void mean_kernel(
    const float* __restrict__ X, const float* __restrict__ W,
    float* __restrict__ out)
{
  int idx = blockIdx.x * blockDim.x + threadIdx.x;
  if (idx >= M_PTS * D_CLS) return;
  int n = idx / D_CLS, d = idx - n * D_CLS;
  const float* xr = X + (long long)n * DIN_;
  float s = 0.f;
  for (int i = 0; i < DIN_; ++i)
    s += xr[i] * W[i * D_CLS + d];
  out[idx] = s;
}

extern "C" void kernel_launch(void* const* d_in, const int* in_sizes, int n_in,
                              void* d_out, int out_size, void* d_ws, size_t ws_size,
                              hipStream_t stream) {
  const float* X  = (const float*)d_in[0];   // (512, 784)
  const float* Jz = (const float*)d_in[1];   // (512, 10, 16384)
  const float* v  = (const float*)d_in[2];   // (16384, 1024)
  const float* Gi = (const float*)d_in[3];   // (1024, 1024)
  const float* W  = (const float*)d_in[4];   // (784, 10)
  float* out = (float*)d_out;                // [mean 5120 | var 51200]

  // workspace: phi (5120x1024) + tmp (5120x1024) = 41,943,040 bytes
  float* phi = (float*)d_ws;
  float* tmp = phi + (size_t)MT * K_DIM;

  dim3 gg(MT / BM, K_DIM / BN);              // (40, 8)
  gemm_wmma_f32<<<gg, 256, 0, stream>>>(Jz, v, phi, S_DIM, 1);
  gemm_wmma_f32<<<gg, 256, 0, stream>>>(phi, Gi, tmp, K_DIM, 0);
  var_diag_kernel<<<M_PTS, 256, 0, stream>>>(phi, tmp, out + MT);
  mean_kernel<<<(MT + 255) / 256, 256, 0, stream>>>(X, W, out);
}